// EXAM_ENCODER_24352464569855
// MI455X (gfx1250) — compile-verified
//
#include <hip/hip_runtime.h>
#include <hip/hip_bf16.h>
#include <math.h>

typedef __attribute__((ext_vector_type(2))) float v2f;
typedef __attribute__((ext_vector_type(8))) float v8f;

#define VSZ   100000
#define EMB   128
#define REGN  7
#define NC    20
#define ML    512
#define BN    64
#define RADN  3
#define ENT   506      // ML - 2*RAD
#define H2    1024     // 2*ML
#define ECHUNK 4

// ---------------------------------------------------------------------------
// Kernel 1: noi partials.  grid (BN, ECHUNK), block 512 = (128 d-lanes x 4 e-slices)
// noi[b,d] = sum_e max_j ( W_region[ti[b,e+3]][j*128+d] * W_word[ti[b,t+r]][d] )
// with m = e*7+j, r = m/506, t = m%506  (faithful to the reference's flat reshape)
// ---------------------------------------------------------------------------
__global__ void __launch_bounds__(512)
noi_partial_kernel(const int* __restrict__ ti,
                   const float* __restrict__ Wr,
                   const float* __restrict__ Ww,
                   float* __restrict__ partial) {
    __shared__ int   s_ti[ML];
    __shared__ float s_red[512];

    const int b     = blockIdx.x;
    const int chunk = blockIdx.y;
    const int tid   = threadIdx.x;
    const int d     = tid & (EMB - 1);   // 0..127
    const int s     = tid >> 7;          // 0..3

    s_ti[tid < ML ? tid : 0] = ti[b * ML + (tid < ML ? tid : 0)];
    __syncthreads();

    const int e0 = chunk * 127;
    const int e1 = (e0 + 127 < ENT) ? (e0 + 127) : ENT;

    float acc = 0.0f;
    for (int e = e0 + s; e < e1; e += 4) {
        const int c = s_ti[e + RADN];
        const float* kr = Wr + (size_t)c * (REGN * EMB) + d;
        int m = e * REGN;
        int r = m / ENT;
        int t = m - r * ENT;
        float mx = -INFINITY;
        #pragma unroll
        for (int j = 0; j < REGN; ++j) {
            const float kv = kr[j * EMB];
            const int   w  = s_ti[t + r];
            const float ev = Ww[(size_t)w * EMB + d];
            mx = fmaxf(mx, kv * ev);
            ++t;
            if (t == ENT) { t = 0; ++r; }
        }
        acc += mx;
    }

    s_red[tid] = acc;
    __syncthreads();
    if (s == 0) {
        float sum = s_red[d] + s_red[d + 128] + s_red[d + 256] + s_red[d + 384];
        partial[((size_t)b * ECHUNK + chunk) * EMB + d] = sum;
    }
}

// ---------------------------------------------------------------------------
// Kernel 2: reduce partials -> noi (deterministic, no atomics)
// ---------------------------------------------------------------------------
__global__ void noi_reduce_kernel(const float* __restrict__ partial,
                                  float* __restrict__ noi) {
    int i = blockIdx.x * blockDim.x + threadIdx.x;   // b*EMB + d
    if (i < BN * EMB) {
        int b = i >> 7;
        int d = i & 127;
        float s = 0.0f;
        #pragma unroll
        for (int c = 0; c < ECHUNK; ++c)
            s += partial[((size_t)b * ECHUNK + c) * EMB + d];
        noi[i] = s;
    }
}

// ---------------------------------------------------------------------------
// Kernel 3: h = relu(noi @ W1 + b1) via V_WMMA_F32_16X16X4_F32.
// M=64 (4 tiles), N=1024 (64 tiles), K=128 (32 wmma steps).
// grid 32 blocks x 256 threads = 8 waves/block; one 16x16 tile per wave.
// ---------------------------------------------------------------------------
__global__ void __launch_bounds__(256)
gemm1_wmma_kernel(const float* __restrict__ noi,
                  const float* __restrict__ W1,
                  const float* __restrict__ b1,
                  float* __restrict__ h) {
    const int wave  = threadIdx.x >> 5;
    const int lane  = threadIdx.x & 31;
    const int l15   = lane & 15;
    const int khalf = (lane >> 4) << 1;                 // 0 or 2
    const int mt    = blockIdx.x >> 3;                  // 0..3
    const int nt    = ((blockIdx.x & 7) << 3) + wave;   // 0..63
    const int row   = mt * 16 + l15;                    // A row for this lane
    const int col   = nt * 16 + l15;                    // B/C col for this lane

    v8f c = {};
    for (int k = 0; k < EMB; k += 4) {
        // A 16x4 f32: lanes 0-15 K=k..k+1, lanes 16-31 K=k+2..k+3 (8B aligned)
        v2f a = *(const v2f*)(noi + row * EMB + k + khalf);
        // B 4x16 f32: mirrored layout, N striped across lanes, K pair per VGPR
        v2f bv;
        bv.x = W1[(k + khalf) * H2 + col];
        bv.y = W1[(k + khalf + 1) * H2 + col];
        c = __builtin_amdgcn_wmma_f32_16x16x4_f32(false, a, false, bv,
                                                  (short)0, c, false, false);
    }

    const float bias  = b1[col];
    const int   rbase = mt * 16 + ((lane >> 4) << 3);   // +8 for upper half-wave
    #pragma unroll
    for (int v = 0; v < 8; ++v) {
        float val = c[v] + bias;
        h[(rbase + v) * H2 + col] = fmaxf(val, 0.0f);
    }
}

// ---------------------------------------------------------------------------
// Kernel 4: logits = h @ W2 + b2 via WMMA.  M=64 (4 tiles), N=20 (pad to 2
// tiles of 16), K=1024 (256 wmma steps).  1 block, 8 waves.  Out-of-range
// columns are masked by multiply (index clamped) so EXEC stays all-ones
// around every WMMA.
// ---------------------------------------------------------------------------
__global__ void __launch_bounds__(256)
gemm2_wmma_kernel(const float* __restrict__ h,
                  const float* __restrict__ W2,
                  const float* __restrict__ b2,
                  float* __restrict__ logits) {
    const int wave  = threadIdx.x >> 5;
    const int lane  = threadIdx.x & 31;
    const int l15   = lane & 15;
    const int khalf = (lane >> 4) << 1;
    const int mt    = wave >> 1;          // 0..3
    const int nt    = wave & 1;           // 0..1
    const int row   = mt * 16 + l15;
    const int col   = nt * 16 + l15;

    const int   ccol  = (col < NC) ? col : 0;
    const float valid = (col < NC) ? 1.0f : 0.0f;

    v8f c = {};
    for (int k = 0; k < H2; k += 4) {
        v2f a = *(const v2f*)(h + row * H2 + k + khalf);
        v2f bv;
        bv.x = W2[(k + khalf) * NC + ccol] * valid;
        bv.y = W2[(k + khalf + 1) * NC + ccol] * valid;
        c = __builtin_amdgcn_wmma_f32_16x16x4_f32(false, a, false, bv,
                                                  (short)0, c, false, false);
    }

    if (col < NC) {
        const float bias  = b2[col];
        const int   rbase = mt * 16 + ((lane >> 4) << 3);
        #pragma unroll
        for (int v = 0; v < 8; ++v)
            logits[(rbase + v) * NC + col] = c[v] + bias;
    }
}

// ---------------------------------------------------------------------------
// Kernel 5: softmax + argmax per row (first-max tie like jnp.argmax).
// d_out layout: [0,1280) logits | [1280,2560) prob | [2560,2624) cls (float)
// ---------------------------------------------------------------------------
__global__ void softmax_argmax_kernel(float* __restrict__ out) {
    int b = blockIdx.x * blockDim.x + threadIdx.x;
    if (b >= BN) return;
    const float* lg  = out + (size_t)b * NC;
    float*       pr  = out + BN * NC + (size_t)b * NC;
    float*       cls = out + 2 * BN * NC;

    float m = lg[0];
    int   am = 0;
    #pragma unroll
    for (int j = 1; j < NC; ++j) {
        float v = lg[j];
        if (v > m) { m = v; am = j; }
    }
    float ex[NC];
    float sum = 0.0f;
    #pragma unroll
    for (int j = 0; j < NC; ++j) { ex[j] = expf(lg[j] - m); sum += ex[j]; }
    float inv = 1.0f / sum;
    #pragma unroll
    for (int j = 0; j < NC; ++j) pr[j] = ex[j] * inv;
    cls[b] = (float)am;
}

// ---------------------------------------------------------------------------
extern "C" void kernel_launch(void* const* d_in, const int* in_sizes, int n_in,
                              void* d_out, int out_size, void* d_ws, size_t ws_size,
                              hipStream_t stream) {
    const int*   ti = (const int*)d_in[0];
    const float* Wr = (const float*)d_in[1];
    const float* Ww = (const float*)d_in[2];
    const float* W1 = (const float*)d_in[3];
    const float* b1 = (const float*)d_in[4];
    const float* W2 = (const float*)d_in[5];
    const float* b2 = (const float*)d_in[6];
    float* out = (float*)d_out;

    float* ws      = (float*)d_ws;
    float* partial = ws;                               // 64*4*128   = 32768 f
    float* noi     = partial + BN * ECHUNK * EMB;      // 64*128     =  8192 f
    float* h       = noi + BN * EMB;                   // 64*1024    = 65536 f
                                                       // total ~416 KB

    noi_partial_kernel<<<dim3(BN, ECHUNK), 512, 0, stream>>>(ti, Wr, Ww, partial);
    noi_reduce_kernel<<<(BN * EMB + 255) / 256, 256, 0, stream>>>(partial, noi);
    gemm1_wmma_kernel<<<32, 256, 0, stream>>>(noi, W1, b1, h);
    gemm2_wmma_kernel<<<1, 256, 0, stream>>>(h, W2, b2, out);
    softmax_argmax_kernel<<<1, 64, 0, stream>>>(out);
}